// NeuralPointCloudTVLoss_5188320494480
// MI455X (gfx1250) — compile-verified
//
#include <hip/hip_runtime.h>
#include <stdint.h>

// Problem constants (match reference)
#define B_    8
#define N_    32768
#define K_    16
#define F_    32
#define EPS_  1e-5f
#define NPTS  (B_ * N_)            // 262144 points
#define PTS_PER_GROUP 8            // 8 waves/block, one wave32 per point (lane == feature)
#define NBLOCKS 2048
#define GROUPS_PER_BLOCK (NPTS / (PTS_PER_GROUP * NBLOCKS))  // 16

// ---------------------------------------------------------------------------
// CDNA5 async global->LDS load (ASYNCcnt-tracked), gfx1250 builtin with
// inline-asm fallback per CDNA5 ISA / 15.18.3.
// ---------------------------------------------------------------------------
typedef __attribute__((address_space(1))) int g1_int;   // global
typedef __attribute__((address_space(3))) int l3_int;   // LDS

__device__ __forceinline__ void async_load_b32(const void* g, void* lds) {
#if __has_builtin(__builtin_amdgcn_global_load_async_to_lds_b32)
  __builtin_amdgcn_global_load_async_to_lds_b32(
      (g1_int*)(void*)g,     // drop const, addrspacecast generic -> global(1)
      (l3_int*)lds,          // addrspacecast generic -> LDS(3)
      /*offset=*/0, /*cpol=*/0);
#else
  uint32_t loff = (uint32_t)(uintptr_t)lds;   // flat low 32 bits == LDS offset
  uint64_t ga   = (uint64_t)(uintptr_t)g;
  asm volatile("global_load_async_to_lds_b32 %0, %1, off"
               :: "v"(loff), "v"(ga) : "memory");
#endif
}

__device__ __forceinline__ void wait_async0() {
#if __has_builtin(__builtin_amdgcn_s_wait_asynccnt)
  __builtin_amdgcn_s_wait_asynccnt(0);
#else
  asm volatile("s_wait_asynccnt 0" ::: "memory");
#endif
}

// Uniform lane broadcast -> SGPR (v_readlane_b32); lane index is a literal.
__device__ __forceinline__ int bcast_i(int v, int l) {
#if __has_builtin(__builtin_amdgcn_readlane)
  return __builtin_amdgcn_readlane(v, l);
#else
  return __shfl(v, l, 32);
#endif
}
__device__ __forceinline__ float bcast_f(float v, int l) {
#if __has_builtin(__builtin_amdgcn_readlane)
  return __int_as_float(__builtin_amdgcn_readlane(__float_as_int(v), l));
#else
  return __shfl(v, l, 32);
#endif
}

// ---------------------------------------------------------------------------
// Main kernel: one wave32 per point, lane = feature index (F == 32 == wave32).
// Phase 1 (lanes 0..15): per-neighbor validity, self-drop, inverse-distance
//   weight c_k from a per-lane (mergeable b96) coord gather.
// Phase 2 (all lanes): readlane-scalarized row index/weight; 16 coalesced
//   128B feature-row gathers in saddr form; local FMA accumulate; ONE
//   shuffle tree per point.
// ---------------------------------------------------------------------------
__global__ __launch_bounds__(256) void tv_kernel(
    const float* __restrict__ coords,       // [B,N,3]
    const float* __restrict__ feats,        // [B,N,F]
    const int*   __restrict__ nidx,         // [B,N,K]
    float*       __restrict__ tv_out,       // [B*N]  (== d_out + 1)
    float*       __restrict__ partials) {   // [NBLOCKS]
  __shared__ float sFeat[PTS_PER_GROUP * F_];  // 256 floats: self feature rows
  __shared__ int   sIdx [PTS_PER_GROUP * K_];  // 128 ints : neighbor indices
  __shared__ float sCrd [PTS_PER_GROUP * 3];   // 24 floats: self coords
  __shared__ float sWave[PTS_PER_GROUP];

  const int tid  = threadIdx.x;
  const int wave = tid >> 5;
  const int lane = tid & 31;
  float blockAcc = 0.f;

  for (int it = 0; it < GROUPS_PER_BLOCK; ++it) {
    const int  group = blockIdx.x * GROUPS_PER_BLOCK + it;
    const long pbase = (long)group * PTS_PER_GROUP;   // first point id in group

    // ---- async stage of this group's self tiles (contiguous in memory) ----
    async_load_b32(feats + pbase * F_ + tid, &sFeat[tid]);            // 1 KB
    if (tid < PTS_PER_GROUP * K_)
      async_load_b32(nidx + pbase * K_ + tid, &sIdx[tid]);            // 512 B
    if (tid < PTS_PER_GROUP * 3)
      async_load_b32(coords + pbase * 3 + tid, &sCrd[tid]);           // 96 B

    // ---- prefetch next group's tiles into cache (global_prefetch_b8) ----
    if (it + 1 < GROUPS_PER_BLOCK) {
      __builtin_prefetch(feats + (pbase + PTS_PER_GROUP) * F_ + tid, 0, 0);
      if (tid < PTS_PER_GROUP * K_)
        __builtin_prefetch(nidx + (pbase + PTS_PER_GROUP) * K_ + tid, 0, 0);
    }

    wait_async0();       // this wave's async loads have landed in LDS
    __syncthreads();     // all waves' staging visible

    const long p = pbase + wave;          // this wave's point id
    const int  b = (int)(p >> 15);        // p / N_   (N_ == 1<<15)
    const int  n = (int)(p & (N_ - 1));   // p % N_
    const size_t rowB = (size_t)b * (size_t)N_;
    const float* __restrict__ featB = feats + rowB * F_;  // batch base (uniform)

    const float selfF = sFeat[wave * F_ + lane];

    // ---- phase 1: lanes 0..15 each own one neighbor slot ----
    int myIdx = (lane < K_) ? sIdx[wave * K_ + lane] : -1;

    // valid-count among the 16 slots (xor masks < 16 keep half-waves separate)
    int nv = (myIdx >= 0) ? 1 : 0;
    nv += __shfl_xor(nv, 8, 32);
    nv += __shfl_xor(nv, 4, 32);
    nv += __shfl_xor(nv, 2, 32);
    nv += __shfl_xor(nv, 1, 32);
    // drop self-neighbor if the point has other valid neighbors
    if (myIdx == n && nv > 1) myIdx = -1;

    int   sidx = 0;     // safe gather index
    float c    = 0.f;   // masked inverse-distance weight
    if (lane < K_) {
      const bool valid = myIdx >= 0;
      sidx = valid ? myIdx : 0;
      const float px = sCrd[wave * 3 + 0];
      const float py = sCrd[wave * 3 + 1];
      const float pz = sCrd[wave * 3 + 2];
      const float* q = coords + (rowB + (size_t)sidx) * 3;  // per-lane b96 gather
      const float dx = q[0] - px, dy = q[1] - py, dz = q[2] - pz;
      const float dist = sqrtf(dx * dx + dy * dy + dz * dz);
      c = valid ? (1.0f / (dist + EPS_)) : 0.f;
    }

    // ---- phase 2: scalarized broadcasts + 16 coalesced row gathers ----
    float lsum = 0.f;
#pragma unroll
    for (int k = 0; k < K_; ++k) {
      const int   si = bcast_i(sidx, k);   // uniform row index (SGPR)
      const float ck = bcast_f(c,    k);   // uniform weight   (SGPR)
      const float fn = featB[(size_t)si * F_ + lane];   // saddr-form 128B row
      lsum = fmaf(ck, fabsf(fn - selfF), lsum);
    }
    // single full-wave reduction per point: tv = sum over features
    lsum += __shfl_xor(lsum, 16, 32);
    lsum += __shfl_xor(lsum,  8, 32);
    lsum += __shfl_xor(lsum,  4, 32);
    lsum += __shfl_xor(lsum,  2, 32);
    lsum += __shfl_xor(lsum,  1, 32);
    const float tv = lsum * 1.0f;   // WEIGHT == 1.0

    if (lane == 0) {
      tv_out[p]  = tv;
      blockAcc  += tv;
    }
    __syncthreads();   // protect LDS tiles before next iteration's staging
  }

  // deterministic per-block partial for the mean
  if (lane == 0) sWave[wave] = blockAcc;
  __syncthreads();
  if (tid == 0) {
    float s = 0.f;
#pragma unroll
    for (int i = 0; i < PTS_PER_GROUP; ++i) s += sWave[i];
    partials[blockIdx.x] = s;
  }
}

// ---------------------------------------------------------------------------
// Final reduction: 2048 partials -> total_loss = sum / (B*N)
// ---------------------------------------------------------------------------
__global__ __launch_bounds__(256) void reduce_kernel(
    const float* __restrict__ partials, float* __restrict__ out) {
  __shared__ float s[256];
  float a = 0.f;
  for (int i = threadIdx.x; i < NBLOCKS; i += 256) a += partials[i];
  s[threadIdx.x] = a;
  __syncthreads();
  for (int off = 128; off > 0; off >>= 1) {
    if (threadIdx.x < off) s[threadIdx.x] += s[threadIdx.x + off];
    __syncthreads();
  }
  if (threadIdx.x == 0) out[0] = s[0] * (1.0f / (float)NPTS);
}

extern "C" void kernel_launch(void* const* d_in, const int* in_sizes, int n_in,
                              void* d_out, int out_size, void* d_ws, size_t ws_size,
                              hipStream_t stream) {
  const float* coords = (const float*)d_in[0];   // [B,N,3] f32
  const float* feats  = (const float*)d_in[1];   // [B,N,F] f32
  const int*   nidx   = (const int*)d_in[2];     // [B,N,K] i32
  float* out = (float*)d_out;                    // [1 + B*N]: loss, tv
  float* ws  = (float*)d_ws;                     // NBLOCKS partial sums (8 KB)

  tv_kernel<<<NBLOCKS, 256, 0, stream>>>(coords, feats, nidx, out + 1, ws);
  reduce_kernel<<<1, 256, 0, stream>>>(ws, out);
}